// HierarchicalAttention_61460982005849
// MI455X (gfx1250) — compile-verified
//
#include <hip/hip_runtime.h>
#include <hip/hip_bf16.h>
#include <math.h>

// ---------------- problem constants ----------------
#define B_    32
#define D_    1024
#define UD_   256
#define ENT_  32
#define NENT_ 64
#define S_    2048   // NENT*ENT
#define C_    65     // NENT+1

typedef float v2f __attribute__((ext_vector_type(2)));
typedef float v4f __attribute__((ext_vector_type(4)));
typedef float v8f __attribute__((ext_vector_type(8)));

__device__ __forceinline__ float waveReduceSum(float v) {
  #pragma unroll
  for (int off = 16; off > 0; off >>= 1) v += __shfl_xor(v, off, 32);
  return v;
}
__device__ __forceinline__ float waveReduceMax(float v) {
  #pragma unroll
  for (int off = 16; off > 0; off >>= 1) v = fmaxf(v, __shfl_xor(v, off, 32));
  return v;
}

// ---------------- k0: zero scratch ----------------
__global__ __launch_bounds__(256) void k0_zero(float* __restrict__ p, int n) {
  int i = blockIdx.x * 256 + threadIdx.x;
  if (i < n) p[i] = 0.0f;
}

// ---------------- k1: q_u = src@Wu^T, q_c = src@Wc^T via f32 WMMA ----------------
// grid (80 n-tiles [16 for Wu, 64 for Wc], 2 m-tiles, 8 k-splits), block 32 (1 wave)
__global__ __launch_bounds__(32)
void k1_qproj(const float* __restrict__ src, const float* __restrict__ Wu,
              const float* __restrict__ Wc, float* __restrict__ qu, float* __restrict__ qc) {
  const int lane  = threadIdx.x;
  const int halfk = lane >> 4;      // 0: K lanes 0-1, 1: K lanes 2-3 (A/B frag layout)
  const int lm    = lane & 15;
  const int nt    = blockIdx.x;
  const int b0    = blockIdx.y * 16;
  const int ks    = blockIdx.z * 128;   // K=1024 split into 8
  const float* W; float* Q; int n0, ldq;
  if (nt < 16) { W = Wu; Q = qu; n0 = nt * 16;        ldq = UD_; }
  else         { W = Wc; Q = qc; n0 = (nt - 16) * 16; ldq = D_;  }
  const float* aptr = src + (size_t)(b0 + lm) * D_;   // A: row = batch m
  const float* bptr = W   + (size_t)(n0 + lm) * D_;   // B: lane = output col n
  v8f acc = {0.f,0.f,0.f,0.f,0.f,0.f,0.f,0.f};
  for (int k = ks; k < ks + 128; k += 4) {
    const int kk = k + 2 * halfk;
    v2f a, b;
    a.x = aptr[kk]; a.y = aptr[kk + 1];
    b.x = bptr[kk]; b.y = bptr[kk + 1];
    acc = __builtin_amdgcn_wmma_f32_16x16x4_f32(false, a, false, b,
                                                (short)0, acc, false, false);
  }
  const int mbase = (lane >= 16) ? 8 : 0;
  #pragma unroll
  for (int r = 0; r < 8; ++r) {
    const int m = r + mbase;
    atomicAdd(&Q[(size_t)(b0 + m) * ldq + n0 + lm], acc[r]);
  }
}

// ---------------- k2: align_units[b,s] = q_u[b] . pos_embs[s,b,:] ----------------
// grid (S/8, B), block 256 (8 waves, one wave per s), q_u[b] cached in LDS.
// pos_embs is a 64MB single-pass stream -> non-temporal loads (don't pollute 192MB L2).
__global__ __launch_bounds__(256)
void k2_units(const float* __restrict__ pe, const float* __restrict__ qu,
              float* __restrict__ au_raw /* (B,S) -> d_out au region */) {
  __shared__ float sq[UD_];
  const int b   = blockIdx.y;
  const int tid = threadIdx.x;
  sq[tid] = qu[b * UD_ + tid];
  __syncthreads();
  const int lane = tid & 31, wid = tid >> 5;
  const int s = blockIdx.x * 8 + wid;
  const v4f* p4 = (const v4f*)(pe + ((size_t)s * B_ + b) * UD_);
  const v4f* q4 = (const v4f*)sq;
  v4f x0 = __builtin_nontemporal_load(p4 + lane);
  v4f x1 = __builtin_nontemporal_load(p4 + 32 + lane);
  v4f y0 = q4[lane];
  v4f y1 = q4[32 + lane];
  float acc = x0.x*y0.x + x0.y*y0.y + x0.z*y0.z + x0.w*y0.w
            + x1.x*y1.x + x1.y*y1.y + x1.z*y1.z + x1.w*y1.w;
  acc = waveReduceSum(acc);
  if (lane == 0) au_raw[(size_t)b * S_ + s] = acc;
}

// ---------------- k3: align_chunks[b,c] = q_c[b] . hl[c,b,:] ----------------
// grid (C, B), block 32
__global__ __launch_bounds__(32)
void k3_chunks(const float* __restrict__ hl, const float* __restrict__ qc,
               float* __restrict__ ac_raw /* (B,C) */) {
  const int c = blockIdx.x, b = blockIdx.y, lane = threadIdx.x;
  const v4f* h4 = (const v4f*)(hl + ((size_t)c * B_ + b) * D_);
  const v4f* q4 = (const v4f*)(qc + (size_t)b * D_);
  float acc = 0.f;
  #pragma unroll 4
  for (int i = lane; i < D_ / 4; i += 32) {
    v4f x = h4[i], y = q4[i];
    acc += x.x*y.x + x.y*y.y + x.z*y.z + x.w*y.w;
  }
  acc = waveReduceSum(acc);
  if (lane == 0) ac_raw[b * C_ + c] = acc;
}

// ---------------- k4a: high-level masked softmax over C, drop chunk 0 ----------------
// grid (B), block 32
__global__ __launch_bounds__(32)
void k4a_hsoft(const float* __restrict__ ac_raw, const int* __restrict__ hmask,
               float* __restrict__ ac_out /* (B, C-1) */) {
  const int b = blockIdx.x, lane = threadIdx.x;
  float v[3];
  float m = -INFINITY;
  #pragma unroll
  for (int t = 0; t < 3; ++t) {
    const int j = lane + t * 32;
    float x = -INFINITY;
    if (j < C_) {
      x = ac_raw[b * C_ + j];
      if (hmask[b * C_ + j] != 0) x = -INFINITY;
    }
    v[t] = x;
    m = fmaxf(m, x);
  }
  m = waveReduceMax(m);
  float sum = 0.f;
  #pragma unroll
  for (int t = 0; t < 3; ++t) {
    const int j = lane + t * 32;
    const float e = (j < C_) ? expf(v[t] - m) : 0.f;
    v[t] = e; sum += e;
  }
  sum = waveReduceSum(sum);
  const float inv = 1.0f / sum;
  #pragma unroll
  for (int t = 0; t < 3; ++t) {
    const int j = lane + t * 32;
    if (j >= 1 && j < C_) ac_out[b * (C_ - 1) + (j - 1)] = v[t] * inv;
  }
}

// ---------------- k4b: per-entity masked softmax (ENT=32 == wave) + combine ----------------
// grid (NENT, B), block 32
__global__ __launch_bounds__(32)
void k4b_lsoft(const int* __restrict__ lmask, const float* __restrict__ ac,
               float* __restrict__ au /* in: raw scores, out: softmax */,
               float* __restrict__ av /* align_vectors */) {
  const int e = blockIdx.x, b = blockIdx.y, lane = threadIdx.x;
  const size_t sidx = (size_t)b * S_ + e * ENT_ + lane;
  float x = au[sidx];
  if (lmask[((size_t)e * B_ + b) * ENT_ + lane] != 0) x = -INFINITY;
  const float m = waveReduceMax(x);     // uniform across wave
  float res;
  if (m == -INFINITY) {                 // all-masked row -> 0 (uniform branch)
    res = 0.0f;
  } else {
    const float ev  = expf(x - m);
    const float sum = waveReduceSum(ev);
    res = ev / sum;
  }
  au[sidx] = res;
  av[sidx] = res * ac[b * (C_ - 1) + e];
}

// ---------------- k5: c[b,:] = sum_s av[b,s] * ll[s,b,:]  (dominant 256MB stream) ----------
// grid (B, 8 s-chunks), block 256 (v4f over full D).
// ll is a single-pass 256MB stream -> non-temporal loads; skip zero-weight rows entirely.
__global__ __launch_bounds__(256)
void k5_ctx(const float* __restrict__ ll, const float* __restrict__ av,
            float* __restrict__ cbuf /* (B,D) accumulated */) {
  const int b   = blockIdx.x;
  const int s0  = blockIdx.y * (S_ / 8);
  const int tid = threadIdx.x;          // v4f index over D
  v4f acc = {0.f, 0.f, 0.f, 0.f};
  for (int s = s0; s < s0 + S_ / 8; ++s) {
    const float w = av[(size_t)b * S_ + s];     // block-uniform
    if (w == 0.0f) continue;                    // masked units: skip the 4KB row read
    const v4f* l4 = (const v4f*)(ll + ((size_t)s * B_ + b) * D_);
    const v4f x = __builtin_nontemporal_load(l4 + tid);
    acc.x += w * x.x; acc.y += w * x.y; acc.z += w * x.z; acc.w += w * x.w;
  }
  float* cb = cbuf + (size_t)b * D_ + tid * 4;
  atomicAdd(cb + 0, acc.x); atomicAdd(cb + 1, acc.y);
  atomicAdd(cb + 2, acc.z); atomicAdd(cb + 3, acc.w);
}

// ---------------- k6: hbuf = [c, src] @ W_out^T via f32 WMMA ----------------
// grid (D/16, 2 m-tiles, 8 k-splits), block 32
__global__ __launch_bounds__(32)
void k6_out(const float* __restrict__ cbuf, const float* __restrict__ src,
            const float* __restrict__ Wout, float* __restrict__ hbuf) {
  const int lane  = threadIdx.x;
  const int halfk = lane >> 4;
  const int lm    = lane & 15;
  const int n0    = blockIdx.x * 16;
  const int b0    = blockIdx.y * 16;
  const int ks    = blockIdx.z * 256;   // K=2048 split into 8 (256-aligned: concat branch uniform)
  const float* bptr = Wout + (size_t)(n0 + lm) * (2 * D_);
  const size_t arow = (size_t)(b0 + lm) * D_;
  v8f acc = {0.f,0.f,0.f,0.f,0.f,0.f,0.f,0.f};
  for (int k = ks; k < ks + 256; k += 4) {
    const int kk = k + 2 * halfk;
    v2f a, b;
    if (kk < D_) { a.x = cbuf[arow + kk];       a.y = cbuf[arow + kk + 1]; }
    else         { a.x = src [arow + kk - D_];  a.y = src [arow + kk - D_ + 1]; }
    b.x = bptr[kk]; b.y = bptr[kk + 1];
    acc = __builtin_amdgcn_wmma_f32_16x16x4_f32(false, a, false, b,
                                                (short)0, acc, false, false);
  }
  const int mbase = (lane >= 16) ? 8 : 0;
  #pragma unroll
  for (int r = 0; r < 8; ++r) {
    const int m = r + mbase;
    atomicAdd(&hbuf[(size_t)(b0 + m) * D_ + n0 + lm], acc[r]);
  }
}

// ---------------- k7: attn_h = tanh(hbuf) ----------------
__global__ __launch_bounds__(256)
void k7_tanh(const float* __restrict__ hbuf, float* __restrict__ out) {
  const int i = blockIdx.x * 256 + threadIdx.x;
  if (i < B_ * D_) out[i] = tanhf(hbuf[i]);
}

// ---------------- launch ----------------
extern "C" void kernel_launch(void* const* d_in, const int* in_sizes, int n_in,
                              void* d_out, int out_size, void* d_ws, size_t ws_size,
                              hipStream_t stream) {
  const float* src   = (const float*)d_in[0];   // (B,D)
  const float* hl    = (const float*)d_in[1];   // (C,B,D)
  const float* pe    = (const float*)d_in[2];   // (S,B,UD)
  const float* ll    = (const float*)d_in[3];   // (S,B,D)
  const int*   lmask = (const int*)d_in[4];     // (NENT,B,ENT) bool
  const int*   hmask = (const int*)d_in[5];     // (1,B,C) bool
  const float* Wu    = (const float*)d_in[6];   // (UD,D)
  const float* Wc    = (const float*)d_in[7];   // (D,D)
  const float* Wo    = (const float*)d_in[8];   // (D,2D)

  float* out    = (float*)d_out;
  float* attn_h = out;                          // (B,D)      32768
  float* av     = out + B_ * D_;                // (B,S)      65536
  float* ac     = av + B_ * S_;                 // (B,C-1)     2048
  float* au     = ac + B_ * (C_ - 1);           // (B,S)      65536

  float* ws    = (float*)d_ws;
  float* qu    = ws;                            // (B,UD)      8192
  float* qc    = qu + B_ * UD_;                 // (B,D)      32768
  float* acraw = qc + B_ * D_;                  // (B,C)       2080
  float* cbuf  = acraw + B_ * C_;               // (B,D)      32768
  float* hbuf  = cbuf + B_ * D_;                // (B,D)      32768
  const int nz = B_ * UD_ + 2 * B_ * D_ + B_ * C_ + B_ * D_;  // zero whole scratch span

  k0_zero  <<<(nz + 255) / 256, 256, 0, stream>>>(ws, nz);
  k1_qproj <<<dim3(80, 2, 8), 32, 0, stream>>>(src, Wu, Wc, qu, qc);
  k2_units <<<dim3(S_ / 8, B_), 256, 0, stream>>>(pe, qu, au);
  k3_chunks<<<dim3(C_, B_), 32, 0, stream>>>(hl, qc, acraw);
  k4a_hsoft<<<B_, 32, 0, stream>>>(acraw, hmask, ac);
  k4b_lsoft<<<dim3(NENT_, B_), 32, 0, stream>>>(lmask, ac, au, av);
  k5_ctx   <<<dim3(B_, 8), 256, 0, stream>>>(ll, av, cbuf);
  k6_out   <<<dim3(D_ / 16, 2, 8), 32, 0, stream>>>(cbuf, src, Wo, hbuf);
  k7_tanh  <<<(B_ * D_ + 255) / 256, 256, 0, stream>>>(hbuf, attn_h);
}